// VADetector_22299470200996
// MI455X (gfx1250) — compile-verified
//
#include <hip/hip_runtime.h>

// Viterbi detector for MI455X (gfx1250, wave32).
//
// One wave per batch element (128 independent waves); state s lives in lane
// s (and is provably re-replicated into lanes 16-31 by every ACS, since all
// cross-lane gathers source lanes 0-15 only). The 16x16 (timestep x state)
// branch-prior tile is computed by V_WMMA_F32_16X16X4_F32:
//   priors_eff[t,s] = y_t * (-sp_s) + 1 * (sp_s^2 / 2)
// The dropped 0.5*y^2 - log(sqrt(2pi)) terms are constant across states for a
// given (b,t), so they shift all path metrics uniformly and cannot change any
// min/argmin decision -> decoded bits are identical to the reference.
//
// Problem is latency-bound (2 MB of input, ~40 MFLOP, strictly sequential
// 4096-step scan): design minimizes LDS round-trips on the per-step
// dependency chain (add -> 2x ds_bpermute -> min).

typedef __attribute__((ext_vector_type(2))) float v2f;
typedef __attribute__((ext_vector_type(8))) float v8f;

#define MEM_LEN 4
#define T_LEN   4096
#define B_LEN   128

__global__ __launch_bounds__(256) void va_detector_kernel(
    const float* __restrict__ y,
    const int*   __restrict__ gammap,
    float*       __restrict__ out) {
  const int lane = threadIdx.x & 31;
  const int wave = (int)((blockIdx.x * blockDim.x + threadIdx.x) >> 5);
  if (wave >= B_LEN) return;
  const int b = wave;
  const float g = (float)(*gammap);
  const bool lo = (lane < 16);

  // state_priors sp[s] for s = lane & 15:
  //   sp[s] = sum_j (1 - 2*bit_j(s)) * exp(-g*j), bit_j = (s >> (3-j)) & 1
  const int s = lane & 15;
  float sp = 0.f;
#pragma unroll
  for (int j = 0; j < MEM_LEN; ++j) {
    float hj  = __expf(-g * (float)j);
    float sym = 1.f - 2.f * (float)((s >> (MEM_LEN - 1 - j)) & 1);
    sp += sym * hj;
  }

  // B fragment (4x16 f32, 2 VGPRs):
  //   VGPR0: lanes 0-15 = row K=0 (-sp_s), lanes 16-31 = row K=2 (0)
  //   VGPR1: lanes 0-15 = row K=1 (sp_s^2/2), lanes 16-31 = row K=3 (0)
  v2f bfrag;
  bfrag.x = lo ? -sp : 0.f;
  bfrag.y = lo ? 0.5f * sp * sp : 0.f;

  float in_prob = 0.f;  // path metric of state (lane&15), replicated per half
  const float* yb = y + (size_t)b * T_LEN;
  float*       ob = out + (size_t)b * T_LEN;

  for (int t0 = 0; t0 < T_LEN; t0 += 16) {
    // Branchless, coalesced tile load: every lane reads yb[t0 + (lane&15)].
    float yv = yb[t0 + s];
    if (t0 + 16 < T_LEN) __builtin_prefetch(yb + t0 + 16, 0, 0);

    // A fragment (16x4 f32, 2 VGPRs):
    //   lanes 0-15: VGPR0 = A[M=lane,K=0] = y[t0+lane], VGPR1 = A[.,K=1] = 1
    //   lanes 16-31: K=2,3 columns, zeroed.
    v2f afrag;
    afrag.x = lo ? yv  : 0.f;
    afrag.y = lo ? 1.f : 0.f;

    // D[r] lanes 0-15  = priors(t0+r,   state=lane)
    // D[r] lanes 16-31 = priors(t0+8+r, state=lane-16)
    v8f d = {};
    d = __builtin_amdgcn_wmma_f32_16x16x4_f32(
        /*neg_a=*/false, afrag, /*neg_b=*/false, bfrag,
        /*c_mod=*/(short)0, d, /*reuse_a=*/false, /*reuse_b=*/false);

    float mybit = 0.f;
#pragma unroll
    for (int r = 0; r < 16; ++r) {
      // (1) decode bit from metrics BEFORE this step's ACS
      //     (in_prob is a correct replica in all 32 lanes)
      float m = in_prob;
      m = fminf(m, __shfl_xor(m, 1));
      m = fminf(m, __shfl_xor(m, 2));
      m = fminf(m, __shfl_xor(m, 4));
      m = fminf(m, __shfl_xor(m, 8));
      unsigned mask = (unsigned)__ballot(in_prob == m) & 0xFFFFu;
      int idx = __ffs((int)mask) - 1;       // first-occurrence argmin
      if (s == r) mybit = (float)(idx & 1); // lane r (and r+16) keeps bit t0+r

      // (2) prior for state s at step r:
      //     r<8 : already resident in lane s of d[r]   (no LDS round-trip)
      //     r>=8: gather from lane s+16 of d[r-8]
      float p = (r < 8) ? d[r] : __shfl(d[r - 8], s + 16);

      // (3) add-compare-select: out[s] = min(tmp[s>>1], tmp[(s>>1)+8])
      //     sources are lanes 0-15 only -> re-replicates in_prob everywhere.
      float tmp = in_prob + p;
      float c0 = __shfl(tmp, s >> 1);
      float c1 = __shfl(tmp, (s >> 1) + 8);
      in_prob = fminf(c0, c1);
    }
    if (lo) ob[t0 + lane] = mybit;
  }
}

extern "C" void kernel_launch(void* const* d_in, const int* in_sizes, int n_in,
                              void* d_out, int out_size, void* d_ws, size_t ws_size,
                              hipStream_t stream) {
  (void)in_sizes; (void)n_in; (void)d_ws; (void)ws_size; (void)out_size;
  const float* y     = (const float*)d_in[0];
  const int*   gamma = (const int*)d_in[1];
  float*       out   = (float*)d_out;
  // 128 waves = 16 blocks x 256 threads (8 wave32s per block)
  va_detector_kernel<<<16, 256, 0, stream>>>(y, gamma, out);
}